// Atomwise_34978213658984
// MI455X (gfx1250) — compile-verified
//
#include <hip/hip_runtime.h>

typedef __attribute__((ext_vector_type(2))) float v2f;
typedef __attribute__((ext_vector_type(4))) float v4f;
typedef __attribute__((ext_vector_type(8))) float v8f;

#define TILE_M    128
#define K_DIM     128
#define HID       64
#define LDS_PITCH 132   // floats; pad so A-fragment reads are bank-conflict-free
#define W1_ELEMS  (K_DIM * HID)   // 8192 floats staged in LDS

__global__ void zero_f32_kernel(float* __restrict__ p, int n) {
    int i = blockIdx.x * blockDim.x + threadIdx.x;
    if (i < n) p[i] = 0.0f;
}

__global__ __launch_bounds__(256)
void atomwise_mlp_kernel(const float* __restrict__ x,
                         const float* __restrict__ W1,
                         const float* __restrict__ b1v,
                         const float* __restrict__ W2,
                         const float* __restrict__ b2v,
                         const int*   __restrict__ idx_m,
                         float* __restrict__ out,
                         int n_atoms)
{
    extern __shared__ float smem[];
    float* xs  = smem;                          // TILE_M * LDS_PITCH floats
    float* w1s = smem + TILE_M * LDS_PITCH;     // W1_ELEMS floats, interleaved

    const int tid        = threadIdx.x;
    const int lane       = tid & 31;
    const int wave       = tid >> 5;                 // 0..7
    const int block_row0 = blockIdx.x * TILE_M;

    // ---- stage W1 into LDS in B-fragment-native layout ----
    // w1s[((k>>2)*64 + n)*4 + (k&3)] = W1[k][n]
    // B-fragment for (k0, n, half) is then a contiguous 8B read.
    #pragma unroll
    for (int j = 0; j < W1_ELEMS / 256; ++j) {
        int e = j * 256 + tid;                       // 0..8191
        int k = e >> 6;                              // coalesced: tid-consecutive n
        int n = e & 63;
        w1s[((k >> 2) * HID + n) * 4 + (k & 3)] = W1[k * HID + n];
    }

    // ---- cooperative, coalesced float4 load of the X tile into LDS ----
    #pragma unroll
    for (int j = 0; j < 16; ++j) {
        int f    = j * 256 + tid;                    // float4 index in tile
        int row  = f >> 5;                           // 32 float4 per row
        int c4   = f & 31;
        int grow = block_row0 + row;
        if (grow >= n_atoms) grow = n_atoms - 1;     // clamp; masked at output
        v4f v = *(const v4f*)(x + (size_t)grow * K_DIM + c4 * 4);
        *(v4f*)(&xs[row * LDS_PITCH + c4 * 4]) = v;
    }
    __syncthreads();

    // ---- WMMA: each wave computes a 16(M) x 64(N) slab of h ----
    const int half = lane >> 4;                      // 0: lanes 0-15, 1: 16-31
    const int l16  = lane & 15;
    const int mrow = wave * 16 + l16;                // A-matrix row (M = lane)

    v8f acc0 = {}, acc1 = {}, acc2 = {}, acc3 = {};

    for (int k0 = 0; k0 < K_DIM; k0 += 4) {
        // A fragment (16x4 f32): VGPR0 = K=kb, VGPR1 = K=kb+1
        v2f a = *(const v2f*)(&xs[mrow * LDS_PITCH + k0 + half * 2]);
        // B fragments (4x16 of W1) from LDS stage; N = l16 + 16*t
        const float* bbase = &w1s[((k0 >> 2) * HID + l16) * 4 + half * 2];
        v2f bf0 = *(const v2f*)(bbase + 0);
        v2f bf1 = *(const v2f*)(bbase + 64);         // +16 n  (256 B)
        v2f bf2 = *(const v2f*)(bbase + 128);        // +32 n
        v2f bf3 = *(const v2f*)(bbase + 192);        // +48 n
        acc0 = __builtin_amdgcn_wmma_f32_16x16x4_f32(false, a, false, bf0, (short)0, acc0, false, false);
        acc1 = __builtin_amdgcn_wmma_f32_16x16x4_f32(false, a, false, bf1, (short)0, acc1, false, false);
        acc2 = __builtin_amdgcn_wmma_f32_16x16x4_f32(false, a, false, bf2, (short)0, acc2, false, false);
        acc3 = __builtin_amdgcn_wmma_f32_16x16x4_f32(false, a, false, bf3, (short)0, acc3, false, false);
    }

    // ---- epilogue: bias + SiLU + second layer (dot with W2[64]) ----
    // C/D layout: n = 16*t + l16 ; m(within wave) = half*8 + r  (r = VGPR index)
    const float bia0 = b1v[l16],      bia1 = b1v[16 + l16];
    const float bia2 = b1v[32 + l16], bia3 = b1v[48 + l16];
    const float w20  = W2[l16],       w21  = W2[16 + l16];
    const float w22  = W2[32 + l16],  w23  = W2[48 + l16];

    float p[8];
    #pragma unroll
    for (int r = 0; r < 8; ++r) {
        float h0 = acc0[r] + bia0; h0 = h0 / (1.0f + __expf(-h0));
        float h1 = acc1[r] + bia1; h1 = h1 / (1.0f + __expf(-h1));
        float h2 = acc2[r] + bia2; h2 = h2 / (1.0f + __expf(-h2));
        float h3 = acc3[r] + bia3; h3 = h3 / (1.0f + __expf(-h3));
        p[r] = h0 * w20 + h1 * w21 + h2 * w22 + h3 * w23;
    }

    // reduce each p[r] over the 16 lanes of this half-wave (n-dimension)
    #pragma unroll
    for (int r = 0; r < 8; ++r) {
        float v = p[r];
        v += __shfl_xor(v, 1);
        v += __shfl_xor(v, 2);
        v += __shfl_xor(v, 4);
        v += __shfl_xor(v, 8);
        p[r] = v;
    }

    if (l16 == 0) {
        const float bout = b2v[0];
        #pragma unroll
        for (int r = 0; r < 8; ++r) {
            int atom = block_row0 + wave * 16 + half * 8 + r;
            if (atom < n_atoms) {
                atomicAdd(&out[idx_m[atom]], p[r] + bout);
            }
        }
    }
}

extern "C" void kernel_launch(void* const* d_in, const int* in_sizes, int n_in,
                              void* d_out, int out_size, void* d_ws, size_t ws_size,
                              hipStream_t stream) {
    const float* x   = (const float*)d_in[0];
    const float* W1  = (const float*)d_in[1];
    const float* b1  = (const float*)d_in[2];
    const float* W2  = (const float*)d_in[3];
    const float* b2  = (const float*)d_in[4];
    const int*   idx = (const int*)d_in[5];
    float* out = (float*)d_out;

    const int n_atoms = in_sizes[0] / K_DIM;

    // segment-sum accumulators must start at zero every call (graph replay safe)
    zero_f32_kernel<<<(out_size + 255) / 256, 256, 0, stream>>>(out, out_size);

    const int grid = (n_atoms + TILE_M - 1) / TILE_M;
    const size_t lds_bytes = (size_t)(TILE_M * LDS_PITCH + W1_ELEMS) * sizeof(float);
    atomwise_mlp_kernel<<<grid, 256, lds_bytes, stream>>>(x, W1, b1, W2, b2, idx,
                                                          out, n_atoms);
}